// GATConv_34514357191304
// MI455X (gfx1250) — compile-verified
//
#include <hip/hip_runtime.h>
#include <hip/hip_bf16.h>
#include <math.h>

#define N_NODES   100000
#define N_EDGES   3200000
#define IN_FEATS  256
#define N_HEADS   8
#define OUT_FEATS 32
#define HF        (N_HEADS * OUT_FEATS)     // 256
#define NH        (N_NODES * N_HEADS)       // 800000
#define EH        (N_EDGES * N_HEADS)       // 25600000
#define N_MTILES  (N_NODES / 16)            // 6250
#define NEG_SLOPE 0.2f

typedef __attribute__((ext_vector_type(2))) float v2f;
typedef __attribute__((ext_vector_type(8))) float v8f;

// ---------------------------------------------------------------------------
// Kernel 0: fold attention vectors into W.
//   elrW[k*16 + h]     = sum_f W[k, h*32+f] * attn_l[h,f]
//   elrW[k*16 + 8 + h] = sum_f W[k, h*32+f] * attn_r[h,f]
// 2048 (k,h) pairs, 32-MAC loop each — trivial.
// ---------------------------------------------------------------------------
__global__ __launch_bounds__(256)
void gat_fold_attn(const float* __restrict__ W,
                   const float* __restrict__ attn_l,
                   const float* __restrict__ attn_r,
                   float* __restrict__ elrW) {
    const int i = blockIdx.x * 256 + threadIdx.x;   // k*8 + h
    if (i >= IN_FEATS * N_HEADS) return;
    const int k = i >> 3;
    const int h = i & 7;
    const float* wrow = W + (size_t)k * HF + h * OUT_FEATS;
    const float* al   = attn_l + h * OUT_FEATS;
    const float* ar   = attn_r + h * OUT_FEATS;
    float sl = 0.f, sr = 0.f;
    #pragma unroll
    for (int f = 0; f < OUT_FEATS; ++f) {
        const float w = wrow[f];
        sl += w * al[f];
        sr += w * ar[f];
    }
    elrW[k * 16 + h]     = sl;
    elrW[k * 16 + 8 + h] = sr;
}

// ---------------------------------------------------------------------------
// Kernel 1: elr = feat @ elrW  ([100000 x 256] x [256 x 16]) via
// V_WMMA_F32_16X16X4_F32. Each wave owns TWO 16-node M-tiles sharing one
// B fragment per K-step -> 2 independent v_wmma per step (no dep chain) and
// half the B traffic per FLOP. elrW (16 KB) is L0-resident (prefetched);
// feat is streamed nontemporally (read exactly once from HBM).
//
// A layout (16x4 MxK): lanes 0-15 -> M=lane, v0=K0,v1=K1; lanes 16-31 ->
//   M=lane-16, v0=K2,v1=K3.  B (4x16 KxN) mirrors with N. C/D (16x16 f32,
//   8 VGPRs): N = lane&15, M = r + 8*(lane>>4).
// ---------------------------------------------------------------------------
__global__ __launch_bounds__(128)
void gat_logits_wmma(const float* __restrict__ feat,
                     const float* __restrict__ elrW,
                     float* __restrict__ elr) {
    const int lane = threadIdx.x & 31;
    const int wave = threadIdx.x >> 5;                 // 0..3
    const int mt0  = (blockIdx.x * 4 + wave) * 2;      // first of 2 M-tiles
    // Warm the 16 KB B table into cache (global_prefetch_b8).
    __builtin_prefetch(elrW + threadIdx.x * 32, 0, 3);
    if (mt0 >= N_MTILES) return;                       // wave-uniform: EXEC stays full

    const int half = lane >> 4;                        // 0 or 1
    const int l16  = lane & 15;
    const int koff = half * 2;                         // lane's K sub-offset

    const float* arow0 = feat + ((size_t)(mt0 * 16) + l16) * IN_FEATS + koff;
    const float* arow1 = arow0 + (size_t)16 * IN_FEATS;
    const float* bcol  = elrW + koff * 16 + l16;

    v8f acc0 = {}; v8f acc1 = {};

    for (int k = 0; k < IN_FEATS; k += 4) {
        v2f a0 = __builtin_nontemporal_load((const v2f*)(arow0 + k));
        v2f a1 = __builtin_nontemporal_load((const v2f*)(arow1 + k));
        v2f b;  b.x = bcol[k * 16];  b.y = bcol[k * 16 + 16];
        acc0 = __builtin_amdgcn_wmma_f32_16x16x4_f32(false, a0, false, b,
                                                     (short)0, acc0, false, false);
        acc1 = __builtin_amdgcn_wmma_f32_16x16x4_f32(false, a1, false, b,
                                                     (short)0, acc1, false, false);
    }

    const int m0 = mt0 * 16 + 8 * half;                // M = r + 8*half
    #pragma unroll
    for (int r = 0; r < 8; ++r) {
        elr[(size_t)(m0 + r) * 16 + l16]       = acc0[r];
        elr[(size_t)(m0 + 16 + r) * 16 + l16]  = acc1[r];
    }
}

// ---------------------------------------------------------------------------
// Kernel 2: init m = -inf, s = 0 (workspace is poisoned by the harness).
// ---------------------------------------------------------------------------
__global__ __launch_bounds__(256)
void gat_init(float* __restrict__ mx, float* __restrict__ sm) {
    const int i = blockIdx.x * 256 + threadIdx.x;
    if (i < NH) {
        mx[i] = -__builtin_huge_valf();
        sm[i] = 0.0f;
    }
}

// e = leaky_relu(el[src,h] + er[dst,h]); elr rows are [el(8) | er(8)].
__device__ __forceinline__ float edge_logit(const int* __restrict__ src,
                                            const int* __restrict__ dst,
                                            const float* __restrict__ elr,
                                            int e, int h, int* d_idx) {
    const int s = src[e];
    const int d = dst[e];
    *d_idx = d * N_HEADS + h;
    float v = elr[s * 16 + h] + elr[d * 16 + 8 + h];
    return v > 0.0f ? v : NEG_SLOPE * v;
}

// ---------------------------------------------------------------------------
// Kernel 3: segment max over destination (global_atomic_max_num_f32 path).
// ---------------------------------------------------------------------------
__global__ __launch_bounds__(256)
void gat_edge_max(const int* __restrict__ src, const int* __restrict__ dst,
                  const float* __restrict__ elr, float* __restrict__ mx) {
    const int i = blockIdx.x * 256 + threadIdx.x;
    if (i >= EH) return;
    int di;
    const float v = edge_logit(src, dst, elr, i >> 3, i & 7, &di);
    __hip_atomic_fetch_max(&mx[di], v, __ATOMIC_RELAXED, __HIP_MEMORY_SCOPE_AGENT);
}

// ---------------------------------------------------------------------------
// Kernel 4: segment sum of exp(e - m[dst]) (recompute e; elr/m are L2-hot).
// ---------------------------------------------------------------------------
__global__ __launch_bounds__(256)
void gat_edge_sum(const int* __restrict__ src, const int* __restrict__ dst,
                  const float* __restrict__ elr, const float* __restrict__ mx,
                  float* __restrict__ sm) {
    const int i = blockIdx.x * 256 + threadIdx.x;
    if (i >= EH) return;
    int di;
    const float v  = edge_logit(src, dst, elr, i >> 3, i & 7, &di);
    const float ez = __expf(v - mx[di]);
    atomicAdd(&sm[di], ez);
}

// ---------------------------------------------------------------------------
// Kernel 5: a[e,h] = exp(e - m[dst]) / s[dst] -> d_out [E,H,1], streamed out
// with nontemporal stores (written once, never re-read).
// ---------------------------------------------------------------------------
__global__ __launch_bounds__(256)
void gat_edge_out(const int* __restrict__ src, const int* __restrict__ dst,
                  const float* __restrict__ elr, const float* __restrict__ mx,
                  const float* __restrict__ sm, float* __restrict__ out) {
    const int i = blockIdx.x * 256 + threadIdx.x;
    if (i >= EH) return;
    int di;
    const float v  = edge_logit(src, dst, elr, i >> 3, i & 7, &di);
    const float a  = __expf(v - mx[di]) / sm[di];
    __builtin_nontemporal_store(a, &out[i]);
}

// ---------------------------------------------------------------------------
extern "C" void kernel_launch(void* const* d_in, const int* in_sizes, int n_in,
                              void* d_out, int out_size, void* d_ws, size_t ws_size,
                              hipStream_t stream) {
    const float* feat   = (const float*)d_in[0];
    const float* W      = (const float*)d_in[1];
    const float* attn_l = (const float*)d_in[2];
    const float* attn_r = (const float*)d_in[3];
    const int*   src    = (const int*)d_in[4];
    const int*   dst    = (const int*)d_in[5];
    float*       out    = (float*)d_out;

    // Workspace layout (floats): elrW (4K) | elr (1.6M) | m (800K) | s (800K)
    float* elrW = (float*)d_ws;
    float* elr  = elrW + IN_FEATS * 16;
    float* mx   = elr + (size_t)N_NODES * 16;
    float* sm   = mx + NH;

    gat_fold_attn  <<<(IN_FEATS * N_HEADS + 255) / 256, 256, 0, stream>>>(W, attn_l, attn_r, elrW);
    gat_logits_wmma<<<(N_MTILES / 2 + 3) / 4, 128, 0, stream>>>(feat, elrW, elr);
    gat_init       <<<(NH + 255) / 256, 256, 0, stream>>>(mx, sm);
    gat_edge_max   <<<(EH + 255) / 256, 256, 0, stream>>>(src, dst, elr, mx);
    gat_edge_sum   <<<(EH + 255) / 256, 256, 0, stream>>>(src, dst, elr, mx, sm);
    gat_edge_out   <<<(EH + 255) / 256, 256, 0, stream>>>(src, dst, elr, mx, sm, out);
}